// SMPL_18519898980452
// MI455X (gfx1250) — compile-verified
//
#include <hip/hip_runtime.h>
#include <math.h>

// ---- problem constants ----
constexpr int cNJ = 24;        // joints
constexpr int cNB = 10;        // betas
constexpr int cNP = 207;       // pose dirs (23*9)
constexpr int cV  = 27578;     // vertices
constexpr int cK  = 24000;     // kept vertices (divisible by 16)
constexpr int cB  = 32;        // batch (2 M-tiles of 16)
constexpr int AK  = 448;       // stacked-A K dim: 208+208+12+12+2+pad(6)

typedef float v2f __attribute__((ext_vector_type(2)));
typedef float v8f __attribute__((ext_vector_type(8)));

__device__ __constant__ int d_par[cNJ] =
    {0,0,0,0,1,2,3,4,5,6,7,8,9,9,9,12,13,14,16,17,18,19,20,21};

// D = A(16x4 f32) x B(4x16 f32) + C(16x16 f32), full fp32 WMMA
#define WMMA_F32(acc, a, b) \
  acc = __builtin_amdgcn_wmma_f32_16x16x4_f32(false, (a), false, (b), (short)0, (acc), false, false)

// ---------------------------------------------------------------------------
// Kernel 1: fold J_regressor against template and shapedirs per gender:
//   jfold[g][j][c][s<10] = sum_v Jreg[g,j,v]*shapedirs[g,v,c,s]
//   jfold[g][j][c][10]   = sum_v Jreg[g,j,v]*v_template[g,v,c]
// ---------------------------------------------------------------------------
__global__ __launch_bounds__(256) void k1_fold(
    const float* __restrict__ jreg, const float* __restrict__ vt,
    const float* __restrict__ sd, float* __restrict__ jfold) {
  int g = blockIdx.x / cNJ, j = blockIdx.x % cNJ;
  int tid = threadIdx.x;
  float acc[33];
#pragma unroll
  for (int i = 0; i < 33; ++i) acc[i] = 0.f;
  for (int v = tid; v < cV; v += 256) {
    float jr = jreg[((size_t)(g * cNJ + j)) * cV + v];
#pragma unroll
    for (int c = 0; c < 3; ++c) {
      acc[c * 11 + 10] += jr * vt[((size_t)(g * cV + v)) * 3 + c];
      const float* sp = sd + ((size_t)(g * cV + v) * 3 + c) * 10;
#pragma unroll
      for (int s = 0; s < 10; ++s) acc[c * 11 + s] += jr * sp[s];
    }
  }
  __shared__ float red[33][256];
#pragma unroll
  for (int i = 0; i < 33; ++i) red[i][tid] = acc[i];
  __syncthreads();
  for (int st = 128; st > 0; st >>= 1) {
    if (tid < st)
      for (int i = 0; i < 33; ++i) red[i][tid] += red[i][tid + st];
    __syncthreads();
  }
  if (tid < 33) jfold[(g * cNJ + j) * 33 + tid] = red[tid][0];
}

// ---------------------------------------------------------------------------
// Kernel 2: one wave, lane = batch. Rodrigues, kinematic chain, and emit:
//   Aws  [32][448]  gender-masked stacked coefficient rows
//   A2ws [12][32][48] gender-masked skinning matrices (3x4 of "results")
//   rootws[32][3]   root joint (== J[b][0])
// ---------------------------------------------------------------------------
__global__ __launch_bounds__(32) void k2_pose(
    const float* __restrict__ pose, const float* __restrict__ betas,
    const float* __restrict__ jfold, float* __restrict__ Aws,
    float* __restrict__ A2ws, float* __restrict__ rootws) {
  int b = threadIdx.x;
  const float* bet = betas + b * 11;
  int g = bet[10] > 0.5f ? 1 : 0;

  __shared__ float Jsh[cB][cNJ][3];
  __shared__ float Gsh[cB][cNJ][12];

  for (int j = 0; j < cNJ; ++j)
#pragma unroll
    for (int c = 0; c < 3; ++c) {
      const float* jf = jfold + ((g * cNJ + j) * 3 + c) * 11;
      float v = jf[10];
#pragma unroll
      for (int s = 0; s < 10; ++s) v += jf[s] * bet[s];
      Jsh[b][j][c] = v;
    }

  // zero our slices (ws is poisoned)
  float* Ar = Aws + b * AK;
  for (int k = 0; k < AK; ++k) Ar[k] = 0.f;
  for (int pq = 0; pq < 12; ++pq)
    for (int k = 0; k < 48; ++k) A2ws[((size_t)pq * cB + b) * 48 + k] = 0.f;

  const int lrbase = g ? 208 : 0;
  for (int j = 0; j < cNJ; ++j) {
    float rx = pose[(b * cNJ + j) * 3 + 0];
    float ry = pose[(b * cNJ + j) * 3 + 1];
    float rz = pose[(b * cNJ + j) * 3 + 2];
    float th = sqrtf(rx * rx + ry * ry + rz * rz);
    th = fmaxf(th, 1.1920929e-7f);
    float sn = sinf(th), cs = cosf(th), oc = 1.f - cs;
    float ux = rx / th, uy = ry / th, uz = rz / th;
    float R[9];
    R[0] = cs + oc * ux * ux; R[1] = oc * ux * uy - sn * uz; R[2] = oc * ux * uz + sn * uy;
    R[3] = oc * uy * ux + sn * uz; R[4] = cs + oc * uy * uy; R[5] = oc * uy * uz - sn * ux;
    R[6] = oc * uz * ux - sn * uy; R[7] = oc * uz * uy + sn * ux; R[8] = cs + oc * uz * uz;
    if (j > 0) {
#pragma unroll
      for (int e = 0; e < 9; ++e) {
        float iv = (e == 0 || e == 4 || e == 8) ? 1.f : 0.f;
        Ar[lrbase + (j - 1) * 9 + e] = R[e] - iv;
      }
    }
    int par = d_par[j];
    float tx = Jsh[b][j][0] - (j ? Jsh[b][par][0] : 0.f);
    float ty = Jsh[b][j][1] - (j ? Jsh[b][par][1] : 0.f);
    float tz = Jsh[b][j][2] - (j ? Jsh[b][par][2] : 0.f);
    float* G = Gsh[b][j];
    if (j == 0) {
      G[0] = R[0]; G[1] = R[1]; G[2]  = R[2]; G[3]  = tx;
      G[4] = R[3]; G[5] = R[4]; G[6]  = R[5]; G[7]  = ty;
      G[8] = R[6]; G[9] = R[7]; G[10] = R[8]; G[11] = tz;
    } else {
      float* P = Gsh[b][par];
#pragma unroll
      for (int p = 0; p < 3; ++p) {
        float a0 = P[p * 4 + 0], a1 = P[p * 4 + 1], a2 = P[p * 4 + 2], a3 = P[p * 4 + 3];
        G[p * 4 + 0] = a0 * R[0] + a1 * R[3] + a2 * R[6];
        G[p * 4 + 1] = a0 * R[1] + a1 * R[4] + a2 * R[7];
        G[p * 4 + 2] = a0 * R[2] + a1 * R[5] + a2 * R[8];
        G[p * 4 + 3] = a0 * tx + a1 * ty + a2 * tz + a3;
      }
    }
  }

  for (int j = 0; j < cNJ; ++j) {
    float jx = Jsh[b][j][0], jy = Jsh[b][j][1], jz = Jsh[b][j][2];
    const float* G = Gsh[b][j];
#pragma unroll
    for (int p = 0; p < 3; ++p) {
      float corr = G[p * 4 + 0] * jx + G[p * 4 + 1] * jy + G[p * 4 + 2] * jz;
#pragma unroll
      for (int q = 0; q < 4; ++q) {
        float m = (q < 3) ? G[p * 4 + q] : (G[p * 4 + 3] - corr);
        A2ws[((size_t)(p * 4 + q) * cB + b) * 48 + (g ? 24 : 0) + j] = m;
      }
    }
  }

  int bb = g ? 428 : 416;
  for (int s = 0; s < 10; ++s) Ar[bb + s] = bet[s];
  Ar[g ? 441 : 440] = 1.f;
  rootws[b * 3 + 0] = Jsh[b][0][0];
  rootws[b * 3 + 1] = Jsh[b][0][1];
  rootws[b * 3 + 2] = Jsh[b][0][2];
}

// ---------------------------------------------------------------------------
// Kernel 3: main WMMA kernel. 4 waves/block, 16 kept vertices per wave,
// all 32 batches (two 16x16 M-tiles) per wave.
//   Phase 1: v_posed via f32 WMMA over stacked K=448 (posedirs/shapedirs/vt,
//            both genders, gender-masked A from LDS).
//   Phase 2: skinning T_pq via 12 f32 WMMA chains (K=48: [w_f | w_m]);
//            A2 fragments come straight from global (73 KB, L2/WGP$-hot) so
//            LDS stays at ~57.7 KB -> 5 blocks/WGP -> 5 waves/SIMD of MLP.
// ---------------------------------------------------------------------------
__global__ __launch_bounds__(128) void k3_skin(
    const float* __restrict__ pd, const float* __restrict__ sd,
    const float* __restrict__ vt, const float* __restrict__ wall,
    const int* __restrict__ keep, const float* __restrict__ ws,
    float* __restrict__ out) {
  extern __shared__ float smem[];
  float* As  = smem;                 // 32*448 = 14336 floats
  float* rts = smem + cB * AK;       // 32*3 roots
  const float* A2g = ws + cB * AK;   // global: 12*32*48 (cache-resident)

  for (int i = threadIdx.x; i < cB * AK; i += 128) smem[i] = ws[i];
  if (threadIdx.x < cB * 3) rts[threadIdx.x] = ws[cB * AK + 12 * cB * 48 + threadIdx.x];
  __syncthreads();

  const int lane = threadIdx.x & 31;
  const int wave = threadIdx.x >> 5;
  const int h  = lane >> 4;        // K-pair selector within fragment
  const int nl = lane & 15;        // column / M-row within tile
  const int n  = (blockIdx.x * 4 + wave) * 16 + nl;  // kept-vertex index
  const int vi = keep[n];

  const v8f zero8 = {0.f, 0.f, 0.f, 0.f, 0.f, 0.f, 0.f, 0.f};
  v8f vp[3][2];
#pragma unroll
  for (int c = 0; c < 3; ++c) { vp[c][0] = zero8; vp[c][1] = zero8; }

  const size_t PDG = (size_t)cV * 3 * cNP;
  const size_t SDG = (size_t)cV * 30;
  const float* pdv[2] = { pd + (size_t)vi * 3 * cNP, pd + PDG + (size_t)vi * 3 * cNP };
  const float* sdv[2] = { sd + (size_t)vi * 30,      sd + SDG + (size_t)vi * 30 };
  const float* Ar0 = As + nl * AK;          // batches 0..15
  const float* Ar1 = As + (16 + nl) * AK;   // batches 16..31

  // warm the long-latency posedirs streams (global_prefetch)
  __builtin_prefetch(pdv[0], 0, 1);
  __builtin_prefetch(pdv[1], 0, 1);

  // ---- posedirs segments (K 0..207 female, 208..415 male; [207],[415]=0) ----
#pragma unroll
  for (int gi = 0; gi < 2; ++gi) {
    const int base = gi ? 208 : 0;
    const float* col = pdv[gi];
    for (int k0 = 0; k0 < 204; k0 += 4) {
      v2f a0 = *(const v2f*)(Ar0 + base + k0 + 2 * h);
      v2f a1 = *(const v2f*)(Ar1 + base + k0 + 2 * h);
#pragma unroll
      for (int c = 0; c < 3; ++c) {
        const float* p = col + c * cNP + k0 + 2 * h;
        v2f bf; bf.x = p[0]; bf.y = p[1];
        WMMA_F32(vp[c][0], a0, bf);
        WMMA_F32(vp[c][1], a1, bf);
      }
    }
    { // remainder: k = 204..206 valid, 207 padded (A==0), masked loads only
      v2f a0 = *(const v2f*)(Ar0 + base + 204 + 2 * h);
      v2f a1 = *(const v2f*)(Ar1 + base + 204 + 2 * h);
#pragma unroll
      for (int c = 0; c < 3; ++c) {
        const float* p = col + c * cNP;
        float x0 = p[204 + 2 * h];   // h=0 -> 204, h=1 -> 206
        float x1 = p[205];
        v2f bf; bf.x = x0; bf.y = h ? 0.f : x1;
        WMMA_F32(vp[c][0], a0, bf);
        WMMA_F32(vp[c][1], a1, bf);
      }
    }
  }

  // ---- shapedirs segments (K 416..427 female, 428..439 male; s 10,11 pad) ----
#pragma unroll
  for (int gi = 0; gi < 2; ++gi) {
    const int base = gi ? 428 : 416;
    const float* col = sdv[gi];
#pragma unroll
    for (int st = 0; st < 2; ++st) {
      v2f a0 = *(const v2f*)(Ar0 + base + 4 * st + 2 * h);
      v2f a1 = *(const v2f*)(Ar1 + base + 4 * st + 2 * h);
#pragma unroll
      for (int c = 0; c < 3; ++c) {
        const float* p = col + c * 10 + 4 * st + 2 * h;
        v2f bf; bf.x = p[0]; bf.y = p[1];
        WMMA_F32(vp[c][0], a0, bf);
        WMMA_F32(vp[c][1], a1, bf);
      }
    }
    { // s = 8,9 valid (h=0); s = 10,11 are zero pad (h=1)
      v2f a0 = *(const v2f*)(Ar0 + base + 8 + 2 * h);
      v2f a1 = *(const v2f*)(Ar1 + base + 8 + 2 * h);
#pragma unroll
      for (int c = 0; c < 3; ++c) {
        const float* p = col + c * 10;
        float x0 = p[8], x1 = p[9];
        v2f bf; bf.x = h ? 0.f : x0; bf.y = h ? 0.f : x1;
        WMMA_F32(vp[c][0], a0, bf);
        WMMA_F32(vp[c][1], a1, bf);
      }
    }
  }

  // ---- template segment (K 440=vt_f, 441=vt_m, 442..443 pad) ----
  {
    v2f a0 = *(const v2f*)(Ar0 + 440 + 2 * h);
    v2f a1 = *(const v2f*)(Ar1 + 440 + 2 * h);
#pragma unroll
    for (int c = 0; c < 3; ++c) {
      float xf = vt[(size_t)vi * 3 + c];
      float xm = vt[(size_t)cV * 3 + (size_t)vi * 3 + c];
      v2f bf; bf.x = h ? 0.f : xf; bf.y = h ? 0.f : xm;
      WMMA_F32(vp[c][0], a0, bf);
      WMMA_F32(vp[c][1], a1, bf);
    }
  }

  // ---- skinning: T_pq = A2[pq] (gender-masked M) x [w_f ; w_m], K = 48 ----
  v2f wfr[12];
  const float* wf = wall + (size_t)vi * cNJ;
  const float* wm = wall + (size_t)cV * cNJ + (size_t)vi * cNJ;
#pragma unroll
  for (int i = 0; i < 12; ++i) {
    const float* p = (4 * i < 24) ? (wf + 4 * i + 2 * h) : (wm + 4 * i - 24 + 2 * h);
    wfr[i].x = p[0]; wfr[i].y = p[1];
  }

  v8f outp[3][2];
#pragma unroll
  for (int p = 0; p < 3; ++p) { outp[p][0] = zero8; outp[p][1] = zero8; }

#pragma unroll
  for (int p = 0; p < 3; ++p) {
#pragma unroll
    for (int q = 0; q < 4; ++q) {
      const int pq = p * 4 + q;
      v8f t0 = zero8, t1 = zero8;
#pragma unroll
      for (int i = 0; i < 12; ++i) {
        v2f a0 = *(const v2f*)(A2g + (pq * cB + nl) * 48 + 4 * i + 2 * h);
        v2f a1 = *(const v2f*)(A2g + (pq * cB + 16 + nl) * 48 + 4 * i + 2 * h);
        WMMA_F32(t0, a0, wfr[i]);
        WMMA_F32(t1, a1, wfr[i]);
      }
      if (q < 3) {
        outp[p][0] += t0 * vp[q][0];
        outp[p][1] += t1 * vp[q][1];
      } else {
        outp[p][0] += t0;
        outp[p][1] += t1;
      }
    }
  }

  // ---- subtract root and store: out[b][n][0..2], c contiguous ----
#pragma unroll
  for (int mt = 0; mt < 2; ++mt) {
#pragma unroll
    for (int r = 0; r < 8; ++r) {
      int b = mt * 16 + r + 8 * h;   // C/D layout: VGPR r -> M = r (+8 for hi half)
      float o0 = outp[0][mt][r] - rts[b * 3 + 0];
      float o1 = outp[1][mt][r] - rts[b * 3 + 1];
      float o2 = outp[2][mt][r] - rts[b * 3 + 2];
      float* dst = out + ((size_t)b * cK + n) * 3;
      dst[0] = o0; dst[1] = o1; dst[2] = o2;
    }
  }
}

// ---------------------------------------------------------------------------
extern "C" void kernel_launch(void* const* d_in, const int* in_sizes, int n_in,
                              void* d_out, int out_size, void* d_ws, size_t ws_size,
                              hipStream_t stream) {
  const float* pose  = (const float*)d_in[0];
  const float* betas = (const float*)d_in[1];
  const float* jreg  = (const float*)d_in[2];
  const float* vt    = (const float*)d_in[3];
  const float* sd    = (const float*)d_in[4];
  const float* pd    = (const float*)d_in[5];
  const float* wall  = (const float*)d_in[6];
  const int*   keep  = (const int*)d_in[7];
  float* out = (float*)d_out;

  float* ws     = (float*)d_ws;
  float* Aws    = ws;                        // 32*448   = 14336 floats
  float* A2ws   = ws + cB * AK;              // 12*32*48 = 18432 floats
  float* rootws = A2ws + 12 * cB * 48;       // 96 floats
  float* jfold  = rootws + cB * 3;           // 2*24*33  = 1584 floats

  k1_fold<<<2 * cNJ, 256, 0, stream>>>(jreg, vt, sd, jfold);
  k2_pose<<<1, 32, 0, stream>>>(pose, betas, jfold, Aws, A2ws, rootws);

  const int ldsBytes = (cB * AK + cB * 3) * 4;  // 57,728 B -> 5 blocks/WGP
  k3_skin<<<cK / 16 / 4, 128, ldsBytes, stream>>>(pd, sd, vt, wall, keep, ws, out);
}